// Rasterizer_35407710388864
// MI455X (gfx1250) — compile-verified
//
#include <hip/hip_runtime.h>
#include <hip/hip_bf16.h>
#include <stdint.h>

typedef __attribute__((ext_vector_type(2))) float v2f;
typedef __attribute__((ext_vector_type(8))) float v8f;

#define NG   1024
#define WID  256
#define HEI  256
#define REC  12      // floats per gaussian record
#define GB   64      // gaussians staged per LDS batch
#define NB   (NG / GB)

// record layout: 0 px, 1 py, 2 conic0, 3 conic1, 4 conic2,
//                5 opac*valid, 6 r, 7 g, 8 b, 9 valid, 10 depth, 11 pad

// ---------------------------------------------------------------- preprocess
__global__ void gs_preprocess(
    const float* __restrict__ means, const float* __restrict__ opac,
    const float* __restrict__ scales, const float* __restrict__ rots,
    const float* __restrict__ shs, const float* __restrict__ view,
    const float* __restrict__ proj, const float* __restrict__ cam,
    const float* __restrict__ tanfx, const float* __restrict__ tanfy,
    float* __restrict__ rec)
{
    int n = blockIdx.x * blockDim.x + threadIdx.x;
    if (n >= NG) return;

    float mx = means[n*3+0], my = means[n*3+1], mz = means[n*3+2];

    // --- quaternion -> rotation, Sigma = (R*S)(R*S)^T
    float qr = rots[n*4+0], qx = rots[n*4+1], qy = rots[n*4+2], qz = rots[n*4+3];
    float qn = rsqrtf(qr*qr + qx*qx + qy*qy + qz*qz);
    qr *= qn; qx *= qn; qy *= qn; qz *= qn;
    float R[3][3] = {
        {1.f-2.f*(qy*qy+qz*qz), 2.f*(qx*qy-qr*qz),     2.f*(qx*qz+qr*qy)},
        {2.f*(qx*qy+qr*qz),     1.f-2.f*(qx*qx+qz*qz), 2.f*(qy*qz-qr*qx)},
        {2.f*(qx*qz-qr*qy),     2.f*(qy*qz+qr*qx),     1.f-2.f*(qx*qx+qy*qy)}};
    float sc[3] = {scales[n*3+0], scales[n*3+1], scales[n*3+2]};
    float Mm[3][3], Sig[3][3];
    for (int i = 0; i < 3; ++i)
        for (int j = 0; j < 3; ++j) Mm[i][j] = R[i][j] * sc[j];
    for (int i = 0; i < 3; ++i)
        for (int k = 0; k < 3; ++k)
            Sig[i][k] = Mm[i][0]*Mm[k][0] + Mm[i][1]*Mm[k][1] + Mm[i][2]*Mm[k][2];

    // --- view transform
    float t0 = view[0]*mx + view[1]*my + view[2]*mz  + view[3];
    float t1 = view[4]*mx + view[5]*my + view[6]*mz  + view[7];
    float t2 = view[8]*mx + view[9]*my + view[10]*mz + view[11];
    bool validz = t2 > 0.2f;
    float tzs = validz ? t2 : 1.0f;

    float tanx = tanfx[0], tany = tanfy[0];
    float fx = (float)WID / (2.0f * tanx);
    float fy = (float)HEI / (2.0f * tany);
    float limx = 1.3f * tanx, limy = 1.3f * tany;
    float tx = fminf(fmaxf(t0 / tzs, -limx), limx) * tzs;
    float ty = fminf(fmaxf(t1 / tzs, -limy), limy) * tzs;

    float J[2][3] = {
        {fx / tzs, 0.f, -fx * tx / (tzs * tzs)},
        {0.f, fy / tzs, -fy * ty / (tzs * tzs)}};
    float T2[2][3];
    for (int i = 0; i < 2; ++i)
        for (int k = 0; k < 3; ++k)
            T2[i][k] = J[i][0]*view[0*4+k] + J[i][1]*view[1*4+k] + J[i][2]*view[2*4+k];
    float tmp[2][3];
    for (int i = 0; i < 2; ++i)
        for (int k = 0; k < 3; ++k)
            tmp[i][k] = T2[i][0]*Sig[0][k] + T2[i][1]*Sig[1][k] + T2[i][2]*Sig[2][k];
    float cov00 = tmp[0][0]*T2[0][0] + tmp[0][1]*T2[0][1] + tmp[0][2]*T2[0][2];
    float cov01 = tmp[0][0]*T2[1][0] + tmp[0][1]*T2[1][1] + tmp[0][2]*T2[1][2];
    float cov11 = tmp[1][0]*T2[1][0] + tmp[1][1]*T2[1][1] + tmp[1][2]*T2[1][2];

    float a  = cov00 + 0.3f;
    float c  = cov11 + 0.3f;
    float bb = cov01;
    float det = a * c - bb * bb;
    bool valid = validz && (det > 0.f);
    float inv = (det > 0.f) ? (1.0f / det) : 0.0f;

    // --- projection to pixel center
    float ph0 = proj[0]*mx  + proj[1]*my  + proj[2]*mz  + proj[3];
    float ph1 = proj[4]*mx  + proj[5]*my  + proj[6]*mz  + proj[7];
    float ph3 = proj[12]*mx + proj[13]*my + proj[14]*mz + proj[15];
    float pw = 1.0f / (ph3 + 1e-7f);
    float px = ((ph0 * pw + 1.0f) * (float)WID - 1.0f) * 0.5f;
    float py = ((ph1 * pw + 1.0f) * (float)HEI - 1.0f) * 0.5f;

    // --- SH -> color
    float dx = mx - cam[0], dy = my - cam[1], dz = mz - cam[2];
    float dn = rsqrtf(dx*dx + dy*dy + dz*dz);
    float x = dx*dn, y = dy*dn, z = dz*dn;
    float xx = x*x, yy = y*y, zz = z*z, xy = x*y, yz = y*z, xz = x*z;
    const float C0 = 0.28209479177387814f;
    const float C1 = 0.4886025119029199f;
    const float C2[5] = {1.0925484305920792f, -1.0925484305920792f, 0.31539156525252005f,
                         -1.0925484305920792f, 0.5462742152960396f};
    const float C3[7] = {-0.5900435899266435f, 2.890611442640554f, -0.4570457994644658f,
                         0.3731763325901154f, -0.4570457994644658f, 1.445305721320277f,
                         -0.5900435899266435f};
    float col[3];
    for (int ch = 0; ch < 3; ++ch) {
        const float* s = shs + n*48 + ch;   // stride 3 per coefficient
        float r = C0 * s[0*3];
        r += -C1*y*s[1*3] + C1*z*s[2*3] - C1*x*s[3*3];
        r += C2[0]*xy*s[4*3] + C2[1]*yz*s[5*3] + C2[2]*(2.f*zz-xx-yy)*s[6*3]
           + C2[3]*xz*s[7*3] + C2[4]*(xx-yy)*s[8*3];
        r += C3[0]*y*(3.f*xx-yy)*s[9*3] + C3[1]*xy*z*s[10*3]
           + C3[2]*y*(4.f*zz-xx-yy)*s[11*3] + C3[3]*z*(2.f*zz-3.f*xx-3.f*yy)*s[12*3]
           + C3[4]*x*(4.f*zz-xx-yy)*s[13*3] + C3[5]*z*(xx-yy)*s[14*3]
           + C3[6]*x*(xx-yy-zz)*s[15*3];
        col[ch] = fmaxf(r + 0.5f, 0.0f);
    }

    float* r = rec + n * REC;
    r[0] = px;  r[1] = py;
    r[2] = c * inv;  r[3] = -bb * inv;  r[4] = a * inv;
    r[5] = valid ? opac[n] : 0.0f;      // validity folded into opacity -> branchless raster
    r[6] = col[0]; r[7] = col[1]; r[8] = col[2];
    r[9] = valid ? 1.0f : 0.0f;
    r[10] = valid ? t2 : 1e10f;
    r[11] = 0.0f;
}

// ---------------------------------------------------------------- depth sort
__global__ void gs_sort_gather(const float* __restrict__ rec, float* __restrict__ srec)
{
    __shared__ float key[NG];
    __shared__ int   ind[NG];
    int tid = threadIdx.x;
    key[tid] = rec[tid * REC + 10];
    ind[tid] = tid;
    __syncthreads();
    for (int k = 2; k <= NG; k <<= 1) {
        for (int j = k >> 1; j > 0; j >>= 1) {
            int ixj = tid ^ j;
            if (ixj > tid) {
                bool up = ((tid & k) == 0);   // ascending block
                float ka = key[tid], kb = key[ixj];
                if ((ka > kb) == up) {
                    key[tid] = kb; key[ixj] = ka;
                    int t = ind[tid]; ind[tid] = ind[ixj]; ind[ixj] = t;
                }
            }
            __syncthreads();
        }
    }
    int src = ind[tid];
    #pragma unroll
    for (int f = 0; f < REC; ++f) srec[tid * REC + f] = rec[src * REC + f];
}

// ---------------------------------------------------------------- rasterize
__device__ __forceinline__ float gauss_alpha(const float* __restrict__ r,
                                             float px_, float py_)
{
    float dx = r[0] - px_;
    float dy = r[1] - py_;
    float power = -0.5f * (r[2]*dx*dx + r[4]*dy*dy) - r[3]*dx*dy;
    float al = fminf(r[5] * __expf(fminf(power, 0.f)), 0.99f);
    // non-short-circuit & -> v_cmp/s_and/cndmask, no EXEC divergence
    int keep = (power <= 0.f) & (al >= (1.0f/255.0f));
    return keep ? al : 0.0f;
}

__global__ __launch_bounds__(256) void gs_raster(
    const float* __restrict__ srec, const float* __restrict__ bg,
    float* __restrict__ out)
{
    __shared__ float sg[2][GB * REC];   // double-buffered batch staging (6 KB)

    int tid  = threadIdx.x;
    int wave = tid >> 5;
    int lane = tid & 31;
    int h    = lane >> 4;           // gaussian-pair half (k=0,1 vs k=2,3)
    int m    = lane & 15;           // pixel-in-wave AND A-matrix row (channel)
    int p    = blockIdx.x * 128 + wave * 16 + m;
    float pixx = (float)(p & (WID - 1));
    float pixy = (float)(p >> 8);

    // branchless color fetch setup: always-valid LDS offset + lane mask
    const int   coff  = 6 + ((m < 3) ? m : 0);     // clamped channel offset
    const float cmask = (m < 3) ? 1.0f : 0.0f;     // zero rows 3..15 of A

    float T = 1.0f;
    v8f acc = {0.f, 0.f, 0.f, 0.f, 0.f, 0.f, 0.f, 0.f};

    unsigned ldsB[2] = { (unsigned)(uintptr_t)(&sg[0][0]),
                         (unsigned)(uintptr_t)(&sg[1][0]) };
    const char* gbase = (const char*)srec;
    const bool copier = (tid < (GB * REC * 4) / 16);   // 192 x B128 per batch

    // prologue: stage batch 0 into buffer 0
    if (copier) {
        unsigned loff = ldsB[0] + (unsigned)tid * 16u;
        unsigned goff = (unsigned)(tid * 16);
        asm volatile("global_load_async_to_lds_b128 %0, %1, %2 offset:0"
                     :: "v"(loff), "v"(goff), "s"(gbase) : "memory");
    }

    for (int b = 0; b < NB; ++b) {
        if (b + 1 < NB) {
            // prefetch next batch into the other buffer, then wait only for
            // the older transfer (async ops from one wave complete in order)
            if (copier) {
                unsigned loff = ldsB[(b + 1) & 1] + (unsigned)tid * 16u;
                unsigned goff = (unsigned)((b + 1) * GB * REC * 4 + tid * 16);
                asm volatile("global_load_async_to_lds_b128 %0, %1, %2 offset:0"
                             :: "v"(loff), "v"(goff), "s"(gbase) : "memory");
            }
            asm volatile("s_wait_asynccnt 1" ::: "memory");
        } else {
            asm volatile("s_wait_asynccnt 0" ::: "memory");
        }
        __syncthreads();

        const float* buf = &sg[b & 1][0];
        #pragma unroll 4
        for (int g4 = 0; g4 < GB; g4 += 4) {
            const float* r0 = &buf[(g4 + 2*h + 0) * REC];
            const float* r1 = &buf[(g4 + 2*h + 1) * REC];
            float a0 = gauss_alpha(r0, pixx, pixy);
            float a1 = gauss_alpha(r1, pixx, pixy);
            float pa    = (1.0f - a0) * (1.0f - a1);
            float other = __shfl_xor(pa, 16, 32);           // swap half-waves
            float prefix = h ? other : 1.0f;                // prod of earlier alphas
            float wEven  = T * prefix * a0;
            float wOdd   = T * prefix * (1.0f - a0) * a1;
            T *= pa * other;                                // full 4-gaussian product

            // A(16x4): row m = channel, cols = 4 sorted gaussians (rows 3..15 zero)
            v2f A, B;
            A.x = r0[coff] * cmask;      // unconditional LDS load, masked by VALU
            A.y = r1[coff] * cmask;
            // B(4x16): front-to-back weights, col = pixel
            B.x = wEven;
            B.y = wOdd;
            // img[channel][pixel] += colors^T * wgt   (the reference einsum)
            acc = __builtin_amdgcn_wmma_f32_16x16x4_f32(
                      false, A, false, B, (short)0, acc, false, false);
        }
        __syncthreads();
    }

    if (h == 0) {   // lanes 0-15 hold C rows 0..2 = R,G,B for pixel p
        out[0 * (WID*HEI) + p] = acc[0] + bg[0] * T;
        out[1 * (WID*HEI) + p] = acc[1] + bg[1] * T;
        out[2 * (WID*HEI) + p] = acc[2] + bg[2] * T;
    }
}

// ---------------------------------------------------------------- launch
extern "C" void kernel_launch(void* const* d_in, const int* in_sizes, int n_in,
                              void* d_out, int out_size, void* d_ws, size_t ws_size,
                              hipStream_t stream)
{
    const float* background = (const float*)d_in[0];
    const float* means3D    = (const float*)d_in[1];
    const float* opacities  = (const float*)d_in[2];
    const float* scales     = (const float*)d_in[3];
    const float* rotations  = (const float*)d_in[4];
    const float* shs        = (const float*)d_in[5];
    const float* view_mat   = (const float*)d_in[6];
    const float* proj_mat   = (const float*)d_in[7];
    const float* cam_pos    = (const float*)d_in[8];
    const float* tan_fovx   = (const float*)d_in[9];
    const float* tan_fovy   = (const float*)d_in[10];
    float* out = (float*)d_out;

    float* rec  = (float*)d_ws;          // NG*REC floats
    float* srec = rec + NG * REC;        // NG*REC floats (depth-sorted)

    gs_preprocess<<<NG / 256, 256, 0, stream>>>(
        means3D, opacities, scales, rotations, shs,
        view_mat, proj_mat, cam_pos, tan_fovx, tan_fovy, rec);

    gs_sort_gather<<<1, NG, 0, stream>>>(rec, srec);

    gs_raster<<<(WID * HEI) / 128, 256, 0, stream>>>(srec, background, out);
}